// KPConv_7842610283225
// MI455X (gfx1250) — compile-verified
//
#include <hip/hip_runtime.h>

typedef __attribute__((ext_vector_type(2))) float v2f;
typedef __attribute__((ext_vector_type(8))) float v8f;

#define KP_EXTENT 0.6f
#define ALPHA 0.3f

constexpr int cB    = 8;
constexpr int cN    = 16384;
constexpr int cP    = 4096;
constexpr int cS    = 32;
constexpr int cK    = 15;     // kernel points
constexpr int cCin  = 64;
constexpr int cCout = 128;
constexpr int TILE_M = 16;    // points per block
constexpr int PITCH  = 964;   // LDS row pitch for A (964 % 64 == 4 -> conflict-free frag loads)
constexpr int KDIM   = cK * cCin; // 960

__global__ __launch_bounds__(256) void kpconv_fused(
    const float* __restrict__ xyz,   // (B,N,3)
    const float* __restrict__ feat,  // (B,N,Cin)
    const int*   __restrict__ qidx,  // (B,P)
    const int*   __restrict__ nidx,  // (B,P,S)
    const float* __restrict__ kpts,  // (K,3)
    const float* __restrict__ W,     // (K,Cin,Cout) viewed as (960,128) row-major
    float*       __restrict__ out)   // (B,P,Cout)
{
    extern __shared__ float smem[];
    float* sA    = smem;                               // 16 x PITCH (weighted, padded)
    float* sInfl = smem + TILE_M * PITCH;              // 4 groups x 32 x 16
    int*   sNid  = (int*)(sInfl + 4 * cS * 16);        // 4 groups x 32
    float* sKP   = (float*)(sNid + 4 * cS);            // 45 floats

    const int tid  = threadIdx.x;
    const int tile = blockIdx.x;
    const int b    = tile / (cP / TILE_M);
    const int p0   = (tile % (cP / TILE_M)) * TILE_M;

    if (tid < cK * 3) sKP[tid] = kpts[tid];
    __syncthreads();

    const int g  = tid >> 6;   // group 0..3 (two waves per group)
    const int lt = tid & 63;

    // ---------------- Phase 1: influence + per-point weighted (VALU) -------
    for (int iter = 0; iter < 4; ++iter) {
        const int m = iter * 4 + g;      // point row within tile
        const int p = p0 + m;

        if (lt < cS) {                   // one fully-active wave per group
            const int s = lt;
            const int q = qidx[b * cP + p];
            const float qx = xyz[((size_t)b * cN + q) * 3 + 0];
            const float qy = xyz[((size_t)b * cN + q) * 3 + 1];
            const float qz = xyz[((size_t)b * cN + q) * 3 + 2];
            const int nn = nidx[((size_t)b * cP + p) * cS + s];
            sNid[g * cS + s] = nn;
            const float rx = xyz[((size_t)b * cN + nn) * 3 + 0] - qx;
            const float ry = xyz[((size_t)b * cN + nn) * 3 + 1] - qy;
            const float rz = xyz[((size_t)b * cN + nn) * 3 + 2] - qz;
            #pragma unroll
            for (int k = 0; k < cK; ++k) {
                const float dx = rx - sKP[k * 3 + 0];
                const float dy = ry - sKP[k * 3 + 1];
                const float dz = rz - sKP[k * 3 + 2];
                const float d  = sqrtf(fmaf(dx, dx, fmaf(dy, dy, dz * dz)));
                float inf = 1.0f - d * (1.0f / KP_EXTENT);
                inf = inf > 0.0f ? inf : 0.0f;
                sInfl[(g * cS + s) * 16 + k] = inf;
            }
        }
        __syncthreads();

        {   // weighted[k][c] = sum_s infl[s][k] * feat[nid[s]][c]; thread = channel c
            const int c = lt;
            float acc[cK];
            #pragma unroll
            for (int k = 0; k < cK; ++k) acc[k] = 0.0f;
            for (int s = 0; s < cS; ++s) {
                const float f = feat[((size_t)b * cN + sNid[g * cS + s]) * cCin + c];
                const float* ir = &sInfl[(g * cS + s) * 16];
                #pragma unroll
                for (int k = 0; k < cK; ++k) acc[k] = fmaf(ir[k], f, acc[k]);
            }
            #pragma unroll
            for (int k = 0; k < cK; ++k) sA[m * PITCH + k * cCin + c] = acc[k];
        }
        __syncthreads();
    }

    // ---------------- Phase 2: (16 x 960) @ (960 x 128) via f32 WMMA -------
    const int w    = tid >> 5;            // wave 0..7 -> 16-column slab of W
    const int lane = tid & 31;
    const int mrow = lane & 15;           // A: M index
    const int koff = (lane >> 4) * 2;     // A/B: lanes<16 hold K 0,1 ; lanes>=16 hold K 2,3
    const int ncol = (w << 4) + (lane & 15);

    v8f acc = {};
    #pragma unroll 4
    for (int kk = 0; kk < KDIM; kk += 4) {
        const float* ap = &sA[mrow * PITCH + kk + koff];
        v2f a; a.x = ap[0]; a.y = ap[1];
        const float* bp = &W[(size_t)(kk + koff) * cCout + ncol];
        v2f bmat; bmat.x = bp[0]; bmat.y = bp[cCout];
        acc = __builtin_amdgcn_wmma_f32_16x16x4_f32(
            /*neg_a=*/false, a, /*neg_b=*/false, bmat,
            /*c_mod=*/(short)0, acc, /*reuse_a=*/false, /*reuse_b=*/false);
    }

    // Epilogue: C/D layout VGPR r -> M=r (lanes 0-15) / M=r+8 (lanes 16-31)
    const int mbase = (lane >> 4) * 8;
    #pragma unroll
    for (int r = 0; r < 8; ++r) {
        float v = acc[r];
        v = v > 0.0f ? v : ALPHA * v;               // leaky relu
        const int m = r + mbase;
        out[((size_t)(b * cP + p0 + m)) * cCout + ncol] = v;
    }
}

extern "C" void kernel_launch(void* const* d_in, const int* in_sizes, int n_in,
                              void* d_out, int out_size, void* d_ws, size_t ws_size,
                              hipStream_t stream) {
    const float* xyz  = (const float*)d_in[0];
    const float* feat = (const float*)d_in[1];
    const int*   qidx = (const int*)d_in[2];
    const int*   nidx = (const int*)d_in[3];
    const float* kpts = (const float*)d_in[4];
    const float* W    = (const float*)d_in[5];
    float* out = (float*)d_out;

    const int blocks = cB * cP / TILE_M; // 2048
    const size_t shmem = (size_t)(TILE_M * PITCH + 4 * cS * 16) * sizeof(float)
                       + (size_t)(4 * cS) * sizeof(int)
                       + 64 * sizeof(float);
    kpconv_fused<<<blocks, 256, shmem, stream>>>(xyz, feat, qidx, nidx, kpts, W, out);
}